// CRF_4672924418652
// MI455X (gfx1250) — compile-verified
//
#include <hip/hip_runtime.h>

#define BSZ 512
#define TLEN 1024
#define NST 64
#define NP 66  // N+2, row stride of A

#define LOG2E 1.4426950408889634f
#define LN2   0.6931471805599453f

typedef __attribute__((ext_vector_type(2))) float v2f;
typedef __attribute__((ext_vector_type(8))) float v8f;

// ---------------------------------------------------------------------------
// Forward recursion: one 128-thread block (4 wave32) per 16 batches.
// Wave w owns output-state tile [16w, 16w+16). alpha kept in log2 domain.
// Per step (ONE barrier, lagged shift, double-buffered LDS):
//   D = exp(A^T) * exp2(alpha2 - s2)   via 16x V_WMMA_F32_16X16X4_F32
//   alpha2' = log2(D) + s2 + P_t*log2e   (masked by t < length)
// ---------------------------------------------------------------------------
__global__ __launch_bounds__(128) void crf_forward_kernel(
    const float* __restrict__ P, const float* __restrict__ A,
    const int* __restrict__ lengths, float* __restrict__ logZ)
{
  const int tid  = threadIdx.x;
  const int w    = tid >> 5;          // wave id == output tile index (0..3)
  const int lane = tid & 31;
  const int bl   = lane & 15;         // batch within this block's group
  const int h    = lane >> 4;         // lane half
  const int b    = blockIdx.x * 16 + bl;
  const int m0   = 16 * w + 8 * h;    // first state held by this lane (D layout)
  const int j    = 16 * w + bl;       // out-state for the A operand (M = lane%16)

  __shared__ __align__(16) float s_exp[2][16 * 68];  // [parity][batch][state]+pad
  __shared__ __align__(16) float s_red[2][64];       // [parity][batch*4 + wave]

  // Constant A operand: expAT[m=j][k=i] = exp(A[i][j]); 16 K-chunks of 4.
  v2f aT[16];
  #pragma unroll
  for (int kc = 0; kc < 16; ++kc) {
    int i = 4 * kc + 2 * h;
    aT[kc].x = __expf(A[(i + 0) * NP + j]);
    aT[kc].y = __expf(A[(i + 1) * NP + j]);
  }
  float aEnd2[8];
  #pragma unroll
  for (int v = 0; v < 8; ++v) aEnd2[v] = A[(m0 + v) * NP + (NST + 1)] * LOG2E;

  const int len = lengths[b];
  const float* Pb = P + (size_t)b * TLEN * NST;

  // alpha2_0 = (A[START, :] + P[:, 0]) * log2e
  v8f alpha;
  {
    float4 pa = *(const float4*)(Pb + m0);
    float4 pb = *(const float4*)(Pb + m0 + 4);
    alpha[0] = (A[NST * NP + m0 + 0] + pa.x) * LOG2E;
    alpha[1] = (A[NST * NP + m0 + 1] + pa.y) * LOG2E;
    alpha[2] = (A[NST * NP + m0 + 2] + pa.z) * LOG2E;
    alpha[3] = (A[NST * NP + m0 + 3] + pa.w) * LOG2E;
    alpha[4] = (A[NST * NP + m0 + 4] + pb.x) * LOG2E;
    alpha[5] = (A[NST * NP + m0 + 5] + pb.y) * LOG2E;
    alpha[6] = (A[NST * NP + m0 + 6] + pb.z) * LOG2E;
    alpha[7] = (A[NST * NP + m0 + 7] + pb.w) * LOG2E;
  }

  // Seed the lagged shift for t=1 (read parity = 1).
  {
    float m = alpha[0];
    #pragma unroll
    for (int v = 1; v < 8; ++v) m = fmaxf(m, alpha[v]);
    m = fmaxf(m, __shfl_xor(m, 16));
    if (lane < 16) s_red[1][bl * 4 + w] = m;
  }
  __syncthreads();

  auto step = [&](int t, const float4 pa, const float4 pb) {
    const int p = t & 1;
    // Lagged per-batch shift (max over alpha_{t-2}, or alpha_0 at t=1, synced last step).
    float4 mm = *(const float4*)&s_red[p][bl * 4];
    float s2 = fmaxf(fmaxf(mm.x, mm.y), fmaxf(mm.z, mm.w));
    // Tile max of the current alpha -> shift for step t+1 (other parity).
    float m = alpha[0];
    #pragma unroll
    for (int v = 1; v < 8; ++v) m = fmaxf(m, alpha[v]);
    m = fmaxf(m, __shfl_xor(m, 16));
    if (lane < 16) s_red[p ^ 1][bl * 4 + w] = m;
    // exp2(alpha - s2) -> LDS (padded rows, conflict-free b128 stores).
    float4 e0, e1;
    e0.x = __builtin_amdgcn_exp2f(alpha[0] - s2);
    e0.y = __builtin_amdgcn_exp2f(alpha[1] - s2);
    e0.z = __builtin_amdgcn_exp2f(alpha[2] - s2);
    e0.w = __builtin_amdgcn_exp2f(alpha[3] - s2);
    e1.x = __builtin_amdgcn_exp2f(alpha[4] - s2);
    e1.y = __builtin_amdgcn_exp2f(alpha[5] - s2);
    e1.z = __builtin_amdgcn_exp2f(alpha[6] - s2);
    e1.w = __builtin_amdgcn_exp2f(alpha[7] - s2);
    *(float4*)&s_exp[p][bl * 68 + m0 + 0] = e0;
    *(float4*)&s_exp[p][bl * 68 + m0 + 4] = e1;
    __syncthreads();
    // D = expAT (16x64) x exp_alpha (64x16), 16 accumulating K=4 WMMAs.
    v8f acc = {0.f, 0.f, 0.f, 0.f, 0.f, 0.f, 0.f, 0.f};
    #pragma unroll
    for (int kc = 0; kc < 16; ++kc) {
      v2f bb = *(const v2f*)&s_exp[p][bl * 68 + 4 * kc + 2 * h];
      acc = __builtin_amdgcn_wmma_f32_16x16x4_f32(
          false, aT[kc], false, bb, (short)0, acc, false, false);
    }
    // alpha2' = log2(D) + s2 + P*log2e, masked per batch by t < length.
    const bool upd = (t < len);
    float pn[8] = {pa.x, pa.y, pa.z, pa.w, pb.x, pb.y, pb.z, pb.w};
    #pragma unroll
    for (int v = 0; v < 8; ++v) {
      float na = __builtin_amdgcn_logf(acc[v]) + fmaf(pn[v], LOG2E, s2);
      alpha[v] = upd ? na : alpha[v];
    }
  };

#define LOADP(u, A4, B4)                                              \
  do {                                                                \
    const float* _pp = Pb + (size_t)(u) * NST + m0;                   \
    A4 = *(const float4*)(_pp);                                       \
    B4 = *(const float4*)(_pp + 4);                                   \
  } while (0)

  // 4-deep register pipeline of P tiles to cover HBM latency.
  float4 a0, b0, a1, b1, a2, b2, a3, b3;
  LOADP(1, a0, b0);
  LOADP(2, a1, b1);
  LOADP(3, a2, b2);
  LOADP(4, a3, b3);

  int t;
  for (t = 1; t + 6 < TLEN; t += 4) {   // t = 1,5,...,1017 (covers 1..1020)
    step(t + 0, a0, b0);  { int u = t + 4;                            LOADP(u, a0, b0); }
    step(t + 1, a1, b1);  { int u = t + 5;                            LOADP(u, a1, b1); }
    step(t + 2, a2, b2);  { int u = t + 6;                            LOADP(u, a2, b2); }
    step(t + 3, a3, b3);  { int u = (t + 7 < TLEN) ? t + 7 : TLEN - 1; LOADP(u, a3, b3); }
  }
  // Tail: t == TLEN-3 here; stages a0..a2 hold P_t for t, t+1, t+2.
  step(t + 0, a0, b0);
  step(t + 1, a1, b1);
  step(t + 2, a2, b2);
#undef LOADP

  // logZ = ln2 * log2sumexp2_j(alpha2_j + aEnd2_j)
  float x[8];
  #pragma unroll
  for (int v = 0; v < 8; ++v) x[v] = alpha[v] + aEnd2[v];
  float xm = x[0];
  #pragma unroll
  for (int v = 1; v < 8; ++v) xm = fmaxf(xm, x[v]);
  xm = fmaxf(xm, __shfl_xor(xm, 16));
  if (lane < 16) s_red[0][bl * 4 + w] = xm;
  __syncthreads();
  float4 g = *(const float4*)&s_red[0][bl * 4];
  float gm = fmaxf(fmaxf(g.x, g.y), fmaxf(g.z, g.w));
  float ss = 0.f;
  #pragma unroll
  for (int v = 0; v < 8; ++v) ss += __builtin_amdgcn_exp2f(x[v] - gm);
  ss += __shfl_xor(ss, 16);
  if (lane < 16) s_red[1][bl * 4 + w] = ss;
  __syncthreads();
  if (tid < 16) {
    float4 q = *(const float4*)&s_red[1][tid * 4];
    logZ[blockIdx.x * 16 + tid] =
        (__builtin_amdgcn_logf(q.x + q.y + q.z + q.w) + gm) * LN2;
  }
}

// ---------------------------------------------------------------------------
// Gold-path score (emissions + transitions + start/end) and final subtraction.
// One block per batch; trivially bandwidth-bound (y: 2 MiB, gathered P: 2 MiB).
// ---------------------------------------------------------------------------
__global__ __launch_bounds__(256) void crf_score_kernel(
    const float* __restrict__ P, const float* __restrict__ A,
    const int* __restrict__ y, const int* __restrict__ lengths,
    const float* __restrict__ logZ, float* __restrict__ out)
{
  const int b   = blockIdx.x;
  const int tid = threadIdx.x;
  const int len = lengths[b];
  const int* yb = y + (size_t)b * TLEN;
  const float* Pb = P + (size_t)b * TLEN * NST;

  float acc = 0.f;
  for (int t = tid; t < len; t += 256) {
    int yt = yb[t];
    acc += Pb[(size_t)t * NST + yt];
    if (t >= 1) acc += A[yb[t - 1] * NP + yt];
  }

  __shared__ float red[256];
  red[tid] = acc;
  __syncthreads();
  #pragma unroll
  for (int s = 128; s > 0; s >>= 1) {
    if (tid < s) red[tid] += red[tid + s];
    __syncthreads();
  }
  if (tid == 0) {
    float score = red[0] + A[NST * NP + yb[0]] + A[yb[len - 1] * NP + (NST + 1)];
    out[b] = logZ[b] - score;
  }
}

extern "C" void kernel_launch(void* const* d_in, const int* in_sizes, int n_in,
                              void* d_out, int out_size, void* d_ws, size_t ws_size,
                              hipStream_t stream) {
  const int*   y   = (const int*)d_in[0];   // (B, T) int32
  const float* P   = (const float*)d_in[1]; // (B, T, N) f32
  const int*   len = (const int*)d_in[2];   // (B,) int32
  const float* A   = (const float*)d_in[3]; // (N+2, N+2) f32
  float* out  = (float*)d_out;              // (B,) f32
  float* logZ = (float*)d_ws;               // B floats of scratch

  crf_forward_kernel<<<BSZ / 16, 128, 0, stream>>>(P, A, len, logZ);
  crf_score_kernel<<<BSZ, 256, 0, stream>>>(P, A, y, len, logZ, out);
}